// QuantizedGroupEmbedding_5952824672397
// MI455X (gfx1250) — compile-verified
//
#include <hip/hip_runtime.h>
#include <stdint.h>

// ---------------------------------------------------------------------------
// Quantized group embedding gather via the CDNA5 Tensor Data Mover (gather
// mode): one TENSOR_LOAD_TO_LDS per block DMAs 8 embedding rows (8 x 8KB)
// into LDS; while the DMA is in flight all waves preload their per-row group
// scales; after the barrier the block dequantizes (int32 -> f32 * scale) out
// of LDS and streams the result to global with nontemporal b128 stores.
// ---------------------------------------------------------------------------

typedef __attribute__((ext_vector_type(4))) unsigned int v4u;
typedef __attribute__((ext_vector_type(8))) int          v8i;
typedef __attribute__((ext_vector_type(4))) int          v4i;
typedef __attribute__((ext_vector_type(4))) float        v4f;

#define DIM            2048
#define GROUPS         16
#define GROUP_SIZE     128          // DIM / GROUPS
#define ROWS_PER_BLOCK 8            // 32-bit-index TDM gather: up to 8 rows
#define THREADS        256
#define ROW_BYTES      (DIM * 4)    // 8 KB per int32 row

__global__ __launch_bounds__(THREADS)
void qembed_tdm_gather_kernel(const int*   __restrict__ weight,
                              const float* __restrict__ scales,
                              const int*   __restrict__ indices,
                              float*       __restrict__ out,
                              int n_tokens,
                              unsigned int vocab)
{
    extern __shared__ char smem_raw[];
    int* lds_rows = (int*)smem_raw;                 // ROWS_PER_BLOCK * DIM int32

    const int base = blockIdx.x * ROWS_PER_BLOCK;
    const int rows = min(ROWS_PER_BLOCK, n_tokens - base);
    if (rows <= 0) return;                          // uniform per block

    // ---- Wave 0 only: build gather-mode D# and issue the TDM DMA. ----
    // Guard is wave-uniform ((tid>>5)==0) so waves 1..7 take a *scalar branch*
    // around the tensor op (TDM issues regardless of EXEC, so EXEC=0 fall-
    // through would over-issue the DMA 8x).
    if ((threadIdx.x >> 5) == 0) {
        unsigned int idx[ROWS_PER_BLOCK];
#pragma unroll
        for (int j = 0; j < ROWS_PER_BLOCK; ++j) {
            const int jj = (j < rows) ? j : (rows - 1);   // pad with last row
            idx[j] = (unsigned int)
                __builtin_amdgcn_readfirstlane((int)indices[base + jj]);
        }

        const uint64_t     gaddr    = (uint64_t)(uintptr_t)weight;
        const unsigned int lds_base = (unsigned int)(uintptr_t)lds_rows;

        // ---- D# group 0 (128b): flags | lds_addr | global_addr | type=2 ----
        v4u g0;
        g0[0] = 0xC0000001u;                    // gather_mode=1, 32b idx, count=1
        g0[1] = lds_base;                       // LDS byte address
        g0[2] = (unsigned int)gaddr;            // global_addr[31:0]
        g0[3] = ((unsigned int)(gaddr >> 32) & 0x01FFFFFFu)  // global_addr[56:32]
              | 0x80000000u;                    // type = 2 ("image")

        // ---- D# group 1 (256b): sizes/strides ----
        v8i g1;
        g1[0] = (int)(2u << 16);                        // data_size=2 (4B), mask=0
        g1[1] = (int)((unsigned int)DIM << 16);         // tensor_dim0[15:0]=2048
        g1[2] = (int)((vocab & 0xFFFFu) << 16);         // tensor_dim1[15:0] | dim0 hi=0
        g1[3] = (int)(((unsigned int)DIM << 16)         // tile_dim0 = 2048
              | ((vocab >> 16) & 0xFFFFu));             // tensor_dim1[31:16]
        g1[4] = rows;                                   // tile_dim1 = #valid indices
        g1[5] = DIM;                                    // tensor_dim0_stride[31:0]
        g1[6] = 0;                                      // stride hi / dim1_stride (ignored)
        g1[7] = 0;

        // ---- D# groups 2+3: 8 x 32-bit row indices ----
        v4i g2, g3;
        g2[0] = (int)idx[0]; g2[1] = (int)idx[1];
        g2[2] = (int)idx[2]; g2[3] = (int)idx[3];
        g3[0] = (int)idx[4]; g3[1] = (int)idx[5];
        g3[2] = (int)idx[6]; g3[3] = (int)idx[7];

        // 6-arg toolchain form: 5th group maps to the (unused) VADDR4 slot.
        v8i g4 = (v8i)0;
        __builtin_amdgcn_tensor_load_to_lds(g0, g1, g2, g3, g4, /*cpol=*/0);
    }

    // ---- All waves: preload per-row group scales while the DMA runs. ----
    // Thread t owns elements [8t, 8t+8) of each row -> single group g = t>>4.
    const int t = threadIdx.x;
    const int g = t >> 4;                       // 8t / GROUP_SIZE
    float srow[ROWS_PER_BLOCK];
#pragma unroll
    for (int r = 0; r < ROWS_PER_BLOCK; ++r) {
        if (r < rows) {
            const int row_idx = indices[base + r];
            srow[r] = scales[(size_t)row_idx * GROUPS + g];
        } else {
            srow[r] = 0.0f;
        }
    }

    // ---- Wave 0: wait for DMA completion, then publish via barrier. ----
    if ((threadIdx.x >> 5) == 0) {
        __builtin_amdgcn_s_wait_tensorcnt(0);   // DMA complete -> LDS valid
    }
    __syncthreads();                            // publish LDS rows to all waves

    // ---- Dequantize out of LDS; stream to global (nontemporal). ----
    const v4i* lrow = (const v4i*)lds_rows;
#pragma unroll
    for (int r = 0; r < ROWS_PER_BLOCK; ++r) {
        if (r >= rows) break;
        const float s = srow[r];

        const v4i w0 = lrow[r * (DIM / 4) + 2 * t];
        const v4i w1 = lrow[r * (DIM / 4) + 2 * t + 1];

        v4f o0, o1;
        o0[0] = (float)w0[0] * s;  o0[1] = (float)w0[1] * s;
        o0[2] = (float)w0[2] * s;  o0[3] = (float)w0[3] * s;
        o1[0] = (float)w1[0] * s;  o1[1] = (float)w1[1] * s;
        o1[2] = (float)w1[2] * s;  o1[3] = (float)w1[3] * s;

        float* outp = out + (size_t)(base + r) * DIM + (size_t)t * 8;
        __builtin_nontemporal_store(o0, (v4f*)outp);
        __builtin_nontemporal_store(o1, (v4f*)(outp + 4));
    }
}

extern "C" void kernel_launch(void* const* d_in, const int* in_sizes, int n_in,
                              void* d_out, int out_size, void* d_ws, size_t ws_size,
                              hipStream_t stream) {
    const int*   weight  = (const int*)d_in[0];    // [VOCAB, DIM] int32 (int8-valued)
    const float* scales  = (const float*)d_in[1];  // [VOCAB, GROUPS] f32
    const int*   indices = (const int*)d_in[2];    // [4, 4096] int
    float*       out     = (float*)d_out;          // [4, 4096, DIM] f32

    const int tokens = in_sizes[2];
    const unsigned int vocab = (unsigned int)(in_sizes[0] / DIM);

    const int blocks = (tokens + ROWS_PER_BLOCK - 1) / ROWS_PER_BLOCK;
    const size_t shmem = (size_t)ROWS_PER_BLOCK * ROW_BYTES;   // 64 KB

    if (blocks > 0) {
        qembed_tdm_gather_kernel<<<blocks, THREADS, shmem, stream>>>(
            weight, scales, indices, out, tokens, vocab);
    }
}